// SparseBlock_66383014527053
// MI455X (gfx1250) — compile-verified
//
#include <hip/hip_runtime.h>

typedef float v2f __attribute__((ext_vector_type(2)));
typedef float v8f __attribute__((ext_vector_type(8)));

#define N_ACTIVE      1000000
#define K_TAPS        27
#define TILES         (N_ACTIVE / 16)            // 62500 full 16-row tiles
#define WAVES_PER_BLK 8
#define BLK_THREADS   256
#define NBLK1         ((TILES + WAVES_PER_BLK - 1) / WAVES_PER_BLK)  // 7813
#define BN_EPS        1e-4f

// ---------------------------------------------------------------------------
// Kernel 1: submanifold sparse conv via V_WMMA_F32_16X16X4_F32
//   - one wave per 16-voxel output tile (all 32 output channels)
//   - weights staged to LDS in B-fragment order (ds_load_b64 per fragment)
//   - emits raw conv output + deterministic per-block BN partial sums
// ---------------------------------------------------------------------------
__global__ __launch_bounds__(BLK_THREADS)
void sconv_wmma_kernel(const float* __restrict__ F,     // [N,32] features
                       const float* __restrict__ W,     // [27,32,32]
                       const int*   __restrict__ NBR,   // [N,27]
                       float*       __restrict__ OUT,   // [N,32] raw conv out
                       float*       __restrict__ PART)  // [NBLK1,64] sum/sumsq
{
  // 27 * 1024 floats = 110592 B  (CDNA5: up to 320KB LDS per workgroup)
  __shared__ float lw[K_TAPS * 1024];
  __shared__ float redS[WAVES_PER_BLK][32];
  __shared__ float redQ[WAVES_PER_BLK][32];

  // Stage weights into LDS, permuted so one ds_load_b64 yields the (K,K+1)
  // pair a lane needs for its B fragment:
  //   lds[k*1024 + (cin>>2)*128 + ((cin>>1)&1)*64 + cout*2 + (cin&1)]
  for (int t = threadIdx.x; t < K_TAPS * 1024; t += BLK_THREADS) {
    int k    = t >> 10;
    int r    = t & 1023;
    int cin  = r >> 5;
    int cout = r & 31;
    lw[k * 1024 + (cin >> 2) * 128 + ((cin >> 1) & 1) * 64 + (cout << 1) + (cin & 1)]
        = W[t];
  }
  __syncthreads();

  const int lane = threadIdx.x & 31;
  const int wave = threadIdx.x >> 5;
  const int m    = lane & 15;   // row within tile / cout within col-tile
  const int half = lane >> 4;   // selects K pair {0,1} vs {2,3} in fragments
  const int tile = blockIdx.x * WAVES_PER_BLK + wave;

  v8f acc0 = {};   // cout 0..15
  v8f acc1 = {};   // cout 16..31

  if (tile < TILES) {
    const int rowbase = tile * 16;
    const int nrow    = rowbase + m;

#pragma unroll 1
    for (int k = 0; k < K_TAPS; ++k) {
      const int nid = NBR[nrow * K_TAPS + k];   // lanes l and l+16 see same nid

      // A fragment (16x4 f32): lane (m,half) holds cols {4c+2*half, 4c+2*half+1}
      v2f A[8];
      if (nid >= 0) {
        const float* p = F + (size_t)nid * 32 + 2 * half;
#pragma unroll
        for (int c = 0; c < 8; ++c)
          A[c] = *(const v2f*)(p + 4 * c);
      } else {
#pragma unroll
        for (int c = 0; c < 8; ++c)
          A[c] = (v2f){0.f, 0.f};
      }

      const float* wk = lw + k * 1024 + half * 64 + (m << 1);
#pragma unroll
      for (int c = 0; c < 8; ++c) {
        v2f b0 = *(const v2f*)(wk + c * 128);        // cout = m
        v2f b1 = *(const v2f*)(wk + c * 128 + 32);   // cout = 16 + m
        acc0 = __builtin_amdgcn_wmma_f32_16x16x4_f32(
            false, A[c], false, b0, (short)0, acc0, false, false);
        acc1 = __builtin_amdgcn_wmma_f32_16x16x4_f32(
            false, A[c], false, b1, (short)0, acc1, false, false);
      }
    }

    // Store raw conv output. C/D layout: VGPR r = row (r + 8*half), lane m = col.
#pragma unroll
    for (int r = 0; r < 8; ++r) {
      const int row = rowbase + r + 8 * half;
      OUT[row * 32 + m]      = acc0[r];
      OUT[row * 32 + 16 + m] = acc1[r];
    }
  }

  // ---- deterministic BN partial sums (no float atomics) ----
  float s0 = 0.f, q0 = 0.f, s1 = 0.f, q1 = 0.f;
#pragma unroll
  for (int r = 0; r < 8; ++r) {
    s0 += acc0[r]; q0 += acc0[r] * acc0[r];
    s1 += acc1[r]; q1 += acc1[r] * acc1[r];
  }
  // fold the two half-wave row groups (rows r and r+8) per channel
  s0 += __shfl_xor(s0, 16, 32);  q0 += __shfl_xor(q0, 16, 32);
  s1 += __shfl_xor(s1, 16, 32);  q1 += __shfl_xor(q1, 16, 32);

  if (lane < 16) {
    redS[wave][m]      = s0;  redS[wave][16 + m] = s1;
    redQ[wave][m]      = q0;  redQ[wave][16 + m] = q1;
  }
  __syncthreads();

  if (wave == 0) {
    float S = 0.f, Q = 0.f;
#pragma unroll
    for (int w = 0; w < WAVES_PER_BLK; ++w) {   // fixed order -> deterministic
      S += redS[w][lane];
      Q += redQ[w][lane];
    }
    PART[(size_t)blockIdx.x * 64 + lane]      = S;
    PART[(size_t)blockIdx.x * 64 + 32 + lane] = Q;
  }
}

// ---------------------------------------------------------------------------
// Kernel 2: fold block partials -> fused BN scale/shift (32 threads)
// ---------------------------------------------------------------------------
__global__ void bn_stats_kernel(const float* __restrict__ PART,
                                const float* __restrict__ gamma,
                                const float* __restrict__ beta,
                                float*       __restrict__ SS,
                                int nblk)
{
  const int c = threadIdx.x;   // 0..31
  float S = 0.f, Q = 0.f;
  for (int b = 0; b < nblk; ++b) {            // fixed order -> deterministic
    S += PART[(size_t)b * 64 + c];
    Q += PART[(size_t)b * 64 + 32 + c];
  }
  const float inv_n = 1.0f / (float)N_ACTIVE;
  const float mean  = S * inv_n;
  const float var   = fmaxf(Q * inv_n - mean * mean, 0.0f);
  const float scale = gamma[c] * rsqrtf(var + BN_EPS);
  SS[c]      = scale;
  SS[32 + c] = beta[c] - mean * scale;
}

// ---------------------------------------------------------------------------
// Kernel 3: in-place normalize + ReLU, float4 vectorized.
// 4*g mod 32 == 4*threadIdx.x mod 32 (blockDim*4 = 1024 ≡ 0 mod 32), so each
// thread's channel quartet is invariant -> scale/shift loaded once.
// ---------------------------------------------------------------------------
__global__ __launch_bounds__(BLK_THREADS)
void bn_relu_kernel(float* __restrict__ OUT, const float* __restrict__ SS)
{
  const size_t g  = (size_t)blockIdx.x * BLK_THREADS + threadIdx.x;
  const int    c0 = (threadIdx.x * 4) & 31;

  float4 sc = *(const float4*)(SS + c0);
  float4 sh = *(const float4*)(SS + 32 + c0);

  float4 v = ((const float4*)OUT)[g];
  v.x = fmaxf(v.x * sc.x + sh.x, 0.0f);
  v.y = fmaxf(v.y * sc.y + sh.y, 0.0f);
  v.z = fmaxf(v.z * sc.z + sh.z, 0.0f);
  v.w = fmaxf(v.w * sc.w + sh.w, 0.0f);
  ((float4*)OUT)[g] = v;
}

// ---------------------------------------------------------------------------
extern "C" void kernel_launch(void* const* d_in, const int* in_sizes, int n_in,
                              void* d_out, int out_size, void* d_ws, size_t ws_size,
                              hipStream_t stream)
{
  const float* features = (const float*)d_in[0];   // [1e6, 32]
  const float* weight   = (const float*)d_in[1];   // [27, 32, 32]
  const float* gamma    = (const float*)d_in[2];   // [32]
  const float* beta     = (const float*)d_in[3];   // [32]
  const int*   nbr      = (const int*)d_in[4];     // [1e6, 27]

  float* out  = (float*)d_out;                     // [1e6, 32]
  float* part = (float*)d_ws;                      // [NBLK1, 64]
  float* ss   = part + (size_t)NBLK1 * 64;         // [64] scale/shift

  sconv_wmma_kernel<<<NBLK1, BLK_THREADS, 0, stream>>>(
      features, weight, nbr, out, part);

  bn_stats_kernel<<<1, 32, 0, stream>>>(part, gamma, beta, ss, NBLK1);

  const int nvec4 = (N_ACTIVE * 32) / 4;           // 8,000,000
  bn_relu_kernel<<<nvec4 / BLK_THREADS, BLK_THREADS, 0, stream>>>(out, ss);
}